// ROIAlign_71571335020610
// MI455X (gfx1250) — compile-verified
//
#include <hip/hip_runtime.h>
#include <stdint.h>

#define OUT_HW 7
#define SAMP 2
#define NSAMP (OUT_HW * SAMP)            // 14 sample coords per axis
#define C_CH 256
#define FH 200
#define FW 200
#define SCALE 0.25f
#define OUT_PER_ROI (C_CH * OUT_HW * OUT_HW)   // 12544 floats
#define OUT_BYTES_PER_ROI (OUT_PER_ROI * 4)    // 50176 bytes
#define N_B128 (OUT_BYTES_PER_ROI / 16)        // 3136 16-byte chunks

// 8-byte pair load with only 4-byte alignment guarantee (x0 can be odd).
struct __attribute__((packed, aligned(4))) f2 { float x, y; };

__global__ __launch_bounds__(256) void roialign_kernel(
    const float* __restrict__ feat, const float* __restrict__ rois,
    float* __restrict__ out)
{
  __shared__ int   s_xb[NSAMP];
  __shared__ float s_wxl[NSAMP], s_wxh[NSAMP];
  __shared__ int   s_y0[NSAMP],  s_y1[NSAMP];
  __shared__ float s_wy0[NSAMP], s_wy1[NSAMP];
  __shared__ float s_out[OUT_PER_ROI];

  const int n = blockIdx.x;
  const int t = threadIdx.x;
  const float* r = rois + (size_t)n * 5;
  const int b = (int)r[0];

  // ---- per-ROI separable geometry: 14 x-entries + 14 y-entries ----
  if (t < 2 * NSAMP) {
    const bool isx = (t < NSAMP);
    const int s = isx ? t : t - NSAMP;
    const int L = isx ? FW : FH;
    const float c1 = r[isx ? 1 : 2] * SCALE;
    const float c2 = r[isx ? 3 : 4] * SCALE;
    const float len = fmaxf(c2 - c1, 1.0f);
    const float pos = c1 + ((float)s + 0.5f) * (len * (1.0f / (float)NSAMP));
    const bool valid = (pos >= -1.0f) && (pos <= (float)L);
    const float pc = fminf(fmaxf(pos, 0.0f), (float)(L - 1));
    const int p0 = (int)pc;                 // pc >= 0, trunc == floor
    const float lp = pc - (float)p0;
    if (isx) {
      const int xb = min(p0, L - 2);        // pair base; swap weights at border
      float wlo = (p0 < L - 1) ? (1.0f - lp) : 0.0f;
      float whi = (p0 < L - 1) ? lp : 1.0f;
      if (!valid) { wlo = 0.0f; whi = 0.0f; }
      s_xb[s] = xb; s_wxl[s] = wlo; s_wxh[s] = whi;
    } else {
      const int yb = min(p0 + 1, L - 1);
      float w0 = 1.0f - lp, w1 = lp;
      if (!valid) { w0 = 0.0f; w1 = 0.0f; }
      s_y0[s] = p0; s_y1[s] = yb; s_wy0[s] = w0; s_wy1[s] = w1;
    }
  }
  __syncthreads();

  // hoist x geometry into registers (shared by every bin row)
  int   xb[NSAMP];
  float wxl[NSAMP], wxh[NSAMP];
#pragma unroll
  for (int s = 0; s < NSAMP; ++s) { xb[s] = s_xb[s]; wxl[s] = s_wxl[s]; wxh[s] = s_wxh[s]; }

  const int c = t;  // one thread per channel
  const float* __restrict__ fbase = feat + ((size_t)(b * C_CH + c)) * (size_t)(FH * FW);

  for (int ph = 0; ph < OUT_HW; ++ph) {
    const int syA = 2 * ph, syB = syA + 1;
    const float wA0 = s_wy0[syA], wA1 = s_wy1[syA];
    const float wB0 = s_wy0[syB], wB1 = s_wy1[syB];
    const float* pA0 = fbase + s_y0[syA] * FW;
    const float* pA1 = fbase + s_y1[syA] * FW;
    const float* pB0 = fbase + s_y0[syB] * FW;
    const float* pB1 = fbase + s_y1[syB] * FW;
    float acc[OUT_HW];
#pragma unroll
    for (int pw = 0; pw < OUT_HW; ++pw) acc[pw] = 0.0f;
#pragma unroll
    for (int pw = 0; pw < OUT_HW; ++pw) {
#pragma unroll
      for (int k = 0; k < SAMP; ++k) {
        const int s = 2 * pw + k;
        const int xo = xb[s];
        const float wl = wxl[s], wh = wxh[s];
        const f2 a0 = *(const f2*)(pA0 + xo);
        const f2 a1 = *(const f2*)(pA1 + xo);
        const f2 b0 = *(const f2*)(pB0 + xo);
        const f2 b1 = *(const f2*)(pB1 + xo);
        acc[pw] += wA0 * (wl * a0.x + wh * a0.y)
                 + wA1 * (wl * a1.x + wh * a1.y)
                 + wB0 * (wl * b0.x + wh * b0.y)
                 + wB1 * (wl * b1.x + wh * b1.y);
      }
    }
#pragma unroll
    for (int pw = 0; pw < OUT_HW; ++pw)
      s_out[c * (OUT_HW * OUT_HW) + ph * OUT_HW + pw] = acc[pw] * 0.25f;
  }

  __syncthreads();

  // ---- CDNA5 async bulk store: LDS -> global, ASYNCcnt-tracked, 16B/lane ----
  const uint64_t obase = (uint64_t)(uintptr_t)(out + (size_t)n * OUT_PER_ROI);
  const uint32_t lbase = (uint32_t)(uintptr_t)(&s_out[0]);  // flat low 32b == LDS addr
  for (int i = t; i < N_B128; i += 256) {
    const uint64_t ga = obase + (uint64_t)((uint32_t)i * 16u);
    const uint32_t la = lbase + (uint32_t)i * 16u;
    asm volatile("global_store_async_from_lds_b128 %0, %1, off"
                 :: "v"(ga), "v"(la) : "memory");
  }
  asm volatile("s_wait_asynccnt 0" ::: "memory");
}

extern "C" void kernel_launch(void* const* d_in, const int* in_sizes, int n_in,
                              void* d_out, int out_size, void* d_ws, size_t ws_size,
                              hipStream_t stream) {
  const float* feat = (const float*)d_in[0];
  const float* rois = (const float*)d_in[1];
  float* out = (float*)d_out;
  const int N = in_sizes[1] / 5;   // 1000 ROIs
  roialign_kernel<<<dim3(N), dim3(256), 0, stream>>>(feat, rois, out);
}